// GraphAttentionLayers_34711925686439
// MI455X (gfx1250) — compile-verified
//
#include <hip/hip_runtime.h>
#include <hip/hip_bf16.h>
#include <math.h>

typedef __attribute__((ext_vector_type(2))) float v2f;
typedef __attribute__((ext_vector_type(8))) float v8f;

// ---------------------------------------------------------------------------
// float atomic max via integer atomics (works for mixed signs, init = -inf)
// ---------------------------------------------------------------------------
__device__ inline void atomicMaxF(float* addr, float val) {
    if (val >= 0.0f) {
        atomicMax((int*)addr, __float_as_int(val));
    } else {
        atomicMin((unsigned int*)addr, __float_as_uint(val));
    }
}

// ---------------------------------------------------------------------------
// fill
// ---------------------------------------------------------------------------
__global__ __launch_bounds__(256) void fill_f32(float* __restrict__ p, float v, long n) {
    long i = (long)blockIdx.x * 256 + threadIdx.x;
    if (i < n) p[i] = v;
}

// ---------------------------------------------------------------------------
// Y[M,128] = X[M,128] @ W[128,128] + bias  using V_WMMA_F32_16X16X4_F32
// block = 128 threads (4 waves). Block tile: 64 rows x 16 cols.
// Each wave computes one 16x16 output tile; K=128 in 32 WMMA steps of K=4.
// ---------------------------------------------------------------------------
__global__ __launch_bounds__(128) void gemm128_wmma(
    const float* __restrict__ X, const float* __restrict__ W,
    const float* __restrict__ bias, float* __restrict__ Y, int M)
{
    __shared__ float As[64][129];   // padded: row stride 129 avoids bank conflicts
    __shared__ float Bs[128][17];

    const int tid   = threadIdx.x;
    const int tileM = blockIdx.x * 64;
    const int tileN = blockIdx.y * 16;

    // stage A: 64x128, float4 coalesced global loads
    for (int j = 0; j < 16; ++j) {
        int li  = j * 512 + tid * 4;
        int r   = li >> 7;
        int col = li & 127;
        int g   = tileM + r;
        float4 v = make_float4(0.f, 0.f, 0.f, 0.f);
        if (g < M) v = *(const float4*)(X + (size_t)g * 128 + col);
        As[r][col + 0] = v.x; As[r][col + 1] = v.y;
        As[r][col + 2] = v.z; As[r][col + 3] = v.w;
    }
    // stage B: 128x16 column block of W
    for (int j = 0; j < 16; ++j) {
        int li = j * 128 + tid;
        int kk = li >> 4;
        int c  = li & 15;
        Bs[kk][c] = W[(size_t)kk * 128 + tileN + c];
    }
    __syncthreads();

    const int wave = tid >> 5;
    const int lane = tid & 31;
    // A 16x4 f32 frag layout: lanes 0-15 -> M=lane, K={k,k+1}; lanes 16-31 -> K={k+2,k+3}
    const int rowA = wave * 16 + (lane & 15);
    const int ksh  = (lane >> 4) * 2;
    const int colB = lane & 15;

    v8f c = {};
    for (int k = 0; k < 128; k += 4) {
        v2f a, b;
        a.x = As[rowA][k + ksh];
        a.y = As[rowA][k + ksh + 1];
        b.x = Bs[k + ksh][colB];
        b.y = Bs[k + ksh + 1][colB];
        c = __builtin_amdgcn_wmma_f32_16x16x4_f32(
                /*neg_a=*/false, a, /*neg_b=*/false, b,
                /*c_mod=*/(short)0, c, /*reuse_a=*/false, /*reuse_b=*/false);
    }

    // C/D layout: VGPR j -> M = j (lanes 0-15) or j+8 (lanes 16-31), N = lane&15
    const float bv    = bias[tileN + colB];
    const int   rbase = tileM + wave * 16 + ((lane >> 4) * 8);
    for (int j = 0; j < 8; ++j) {
        int r = rbase + j;
        if (r < M) Y[(size_t)r * 128 + tileN + colB] = c[j] + bv;
    }
}

// ---------------------------------------------------------------------------
// per-edge attention logits: alpha[e,h] = sum_c att[h,c]*leaky_relu(xl[s,h,c]+xr[d,h,c])
// one wave per edge; atomic segment-max into m[dst,h]
// ---------------------------------------------------------------------------
__global__ __launch_bounds__(256) void edge_alpha_k(
    const float* __restrict__ xl, const float* __restrict__ xr,
    const float* __restrict__ att, const int* __restrict__ src,
    const int* __restrict__ dst, float* __restrict__ alpha,
    float* __restrict__ mx, int Ee)
{
    int lane = threadIdx.x & 31;
    int wv   = threadIdx.x >> 5;
    long e   = (long)blockIdx.x * 8 + wv;
    if (e >= Ee) return;
    int s = src[e], d = dst[e];
    const float* pl = xl + (size_t)s * 128;
    const float* pr = xr + (size_t)d * 128;
    for (int h = 0; h < 4; ++h) {
        float v = pl[h * 32 + lane] + pr[h * 32 + lane];
        v = (v > 0.f) ? v : 0.2f * v;               // leaky_relu slope 0.2
        float t = v * att[h * 32 + lane];
        for (int o = 16; o; o >>= 1) t += __shfl_xor(t, o, 32);
        if (lane == 0) {
            alpha[e * 4 + h] = t;
            atomicMaxF(&mx[(size_t)d * 4 + h], t);
        }
    }
}

// exp(alpha - m[dst]) and segment-sum into den
__global__ __launch_bounds__(256) void edge_exp_k(
    float* __restrict__ alpha, const float* __restrict__ mx,
    float* __restrict__ den, const int* __restrict__ dst, int Ee)
{
    long i = (long)blockIdx.x * 256 + threadIdx.x;
    if (i >= (long)Ee * 4) return;
    long e = i >> 2;
    int  h = (int)(i & 3);
    int  d = dst[e];
    float a = __expf(alpha[i] - mx[(size_t)d * 4 + h]);
    alpha[i] = a;
    atomicAdd(&den[(size_t)d * 4 + h], a);
}

// weighted scatter: acc[dst,h,c] += xl[src,h,c] * a/(den+1e-16)
__global__ __launch_bounds__(256) void edge_scatter_k(
    const float* __restrict__ xl, const float* __restrict__ alpha,
    const float* __restrict__ den, const int* __restrict__ src,
    const int* __restrict__ dst, float* __restrict__ acc, int Ee)
{
    int lane = threadIdx.x & 31;
    int wv   = threadIdx.x >> 5;
    long e   = (long)blockIdx.x * 8 + wv;
    if (e >= Ee) return;
    int s = src[e], d = dst[e];
    for (int h = 0; h < 4; ++h) {
        float w = alpha[e * 4 + h] / (den[(size_t)d * 4 + h] + 1e-16f);
        atomicAdd(&acc[(size_t)d * 128 + h * 32 + lane],
                  xl[(size_t)s * 128 + h * 32 + lane] * w);
    }
}

// ---------------------------------------------------------------------------
// post: +bias, LayerNorm(128), ReLU, optional residual. one wave per node.
// ---------------------------------------------------------------------------
__global__ __launch_bounds__(256) void post_concat_k(
    const float* __restrict__ acc, const float* __restrict__ bias,
    const float* __restrict__ g, const float* __restrict__ b,
    const float* __restrict__ res, float* __restrict__ out, int Nn)
{
    int lane = threadIdx.x & 31;
    int wv   = threadIdx.x >> 5;
    long n   = (long)blockIdx.x * 8 + wv;
    if (n >= Nn) return;
    const float* a = acc + (size_t)n * 128;
    float v[4];
    float s = 0.f;
    for (int j = 0; j < 4; ++j) { v[j] = a[j * 32 + lane] + bias[j * 32 + lane]; s += v[j]; }
    for (int o = 16; o; o >>= 1) s += __shfl_xor(s, o, 32);
    float mu = s * (1.0f / 128.0f);
    float ss = 0.f;
    for (int j = 0; j < 4; ++j) { float d = v[j] - mu; ss += d * d; }
    for (int o = 16; o; o >>= 1) ss += __shfl_xor(ss, o, 32);
    float inv = rsqrtf(ss * (1.0f / 128.0f) + 1e-5f);
    for (int j = 0; j < 4; ++j) {
        float h = (v[j] - mu) * inv * g[j * 32 + lane] + b[j * 32 + lane];
        h = fmaxf(h, 0.f);                              // relu first ...
        if (res) h += res[(size_t)n * 128 + j * 32 + lane]; // ... then residual
        out[(size_t)n * 128 + j * 32 + lane] = h;
    }
}

// final layer: head-mean -> [N,32], LN over 32, ReLU. lane == channel.
__global__ __launch_bounds__(256) void post_final_k(
    const float* __restrict__ acc, const float* __restrict__ bias,
    const float* __restrict__ g, const float* __restrict__ b,
    float* __restrict__ out, int Nn)
{
    int lane = threadIdx.x & 31;
    int wv   = threadIdx.x >> 5;
    long n   = (long)blockIdx.x * 8 + wv;
    if (n >= Nn) return;
    const float* a = acc + (size_t)n * 128;
    float v = 0.25f * (a[lane] + a[32 + lane] + a[64 + lane] + a[96 + lane]) + bias[lane];
    float s = v;
    for (int o = 16; o; o >>= 1) s += __shfl_xor(s, o, 32);
    float mu = s * (1.0f / 32.0f);
    float d  = v - mu;
    float ss = d * d;
    for (int o = 16; o; o >>= 1) ss += __shfl_xor(ss, o, 32);
    float inv = rsqrtf(ss * (1.0f / 32.0f) + 1e-5f);
    out[(size_t)n * 32 + lane] = fmaxf((v - mu) * inv * g[lane] + b[lane], 0.f);
}

// ---------------------------------------------------------------------------
extern "C" void kernel_launch(void* const* d_in, const int* in_sizes, int n_in,
                              void* d_out, int out_size, void* d_ws, size_t ws_size,
                              hipStream_t stream)
{
    const float* x    = (const float*)d_in[0];
    const int*   ei   = (const int*)d_in[1];
    const float* Wl   = (const float*)d_in[2];
    const float* bl   = (const float*)d_in[3];
    const float* Wr   = (const float*)d_in[4];
    const float* br   = (const float*)d_in[5];
    const float* att  = (const float*)d_in[6];
    const float* ob01 = (const float*)d_in[7];
    const float* ob2  = (const float*)d_in[8];
    const float* g01  = (const float*)d_in[9];
    const float* b01  = (const float*)d_in[10];
    const float* g2   = (const float*)d_in[11];
    const float* b2   = (const float*)d_in[12];

    const int Nn = in_sizes[0] / 128;
    const int Ee = in_sizes[1] / 2;
    const int* src = ei;
    const int* dst = ei + Ee;

    float* ws = (float*)d_ws;
    const size_t nf = (size_t)Nn * 128;
    float* xl    = ws; ws += nf;
    float* xr    = ws; ws += nf;
    float* acc   = ws; ws += nf;
    float* x1    = ws; ws += nf;
    float* x2    = ws; ws += nf;
    float* alpha = ws; ws += (size_t)Ee * 4;
    float* mbuf  = ws; ws += (size_t)Nn * 4;
    float* den   = ws; ws += (size_t)Nn * 4;

    dim3 gemmGrid((Nn + 63) / 64, 8);
    const int edgeBlocks  = (Ee + 7) / 8;          // 8 waves (edges) per block
    const int nodeBlocks  = (Nn + 7) / 8;          // 8 waves (nodes) per block
    const int fillNH      = (Nn * 4 + 255) / 256;
    const int fillAcc     = (int)((nf + 255) / 256);
    const int expBlocks   = (int)(((long)Ee * 4 + 255) / 256);

    const float* xin = x;
    for (int L = 0; L < 3; ++L) {
        fill_f32<<<fillNH, 256, 0, stream>>>(mbuf, -INFINITY, (long)Nn * 4);
        fill_f32<<<fillNH, 256, 0, stream>>>(den, 0.0f, (long)Nn * 4);
        fill_f32<<<fillAcc, 256, 0, stream>>>(acc, 0.0f, (long)nf);

        gemm128_wmma<<<gemmGrid, 128, 0, stream>>>(xin, Wl + (size_t)L * 16384,
                                                   bl + L * 128, xl, Nn);
        gemm128_wmma<<<gemmGrid, 128, 0, stream>>>(xin, Wr + (size_t)L * 16384,
                                                   br + L * 128, xr, Nn);

        edge_alpha_k<<<edgeBlocks, 256, 0, stream>>>(xl, xr, att + L * 128,
                                                     src, dst, alpha, mbuf, Ee);
        edge_exp_k<<<expBlocks, 256, 0, stream>>>(alpha, mbuf, den, dst, Ee);
        edge_scatter_k<<<edgeBlocks, 256, 0, stream>>>(xl, alpha, den, src, dst, acc, Ee);

        if (L < 2) {
            float* xout      = (L == 0) ? x1 : x2;
            const float* res = (L == 1) ? x1 : nullptr;
            post_concat_k<<<nodeBlocks, 256, 0, stream>>>(acc, ob01 + L * 128,
                                                          g01 + L * 128, b01 + L * 128,
                                                          res, xout, Nn);
            xin = xout;
        } else {
            post_final_k<<<nodeBlocks, 256, 0, stream>>>(acc, ob2, g2, b2,
                                                         (float*)d_out, Nn);
        }
    }
}